// LowRankDirectedKernelOnFeatures_73014444032772
// MI455X (gfx1250) — compile-verified
//
#include <hip/hip_runtime.h>
#include <math.h>

typedef float v2f __attribute__((ext_vector_type(2)));
typedef float v4f __attribute__((ext_vector_type(4)));
typedef float v8f __attribute__((ext_vector_type(8)));

#define BATCH 32
#define NDIM  2048
#define DDIM  64
#define RDIM  32
#define TDIM  31
#define ODIM  12
#define HIDD  128

__device__ __forceinline__ float softplus_f(float x) {
    return fmaxf(x, 0.0f) + log1pf(expf(-fabsf(x)));
}

// ---------------- Kernel 1: elementwise softplus (P and Q) ----------------
__global__ void sp_kernel(const float* __restrict__ in, float* __restrict__ out, int n) {
    int i = blockIdx.x * blockDim.x + threadIdx.x;
    if (i < n) out[i] = softplus_f(in[i]);
}

// ---------------- Kernel 2: U[b] = Q^T (32x2048) @ H[b] (2048x64) ---------
// One wave per (batch, 16x16 tile). 2 r-tiles x 4 d-tiles = 8 tiles/batch.
__global__ __launch_bounds__(32) void u_kernel(const float* __restrict__ H,
                                               const float* __restrict__ Q,
                                               float* __restrict__ U) {
    int blk  = blockIdx.x;          // 0..255
    int tile = blk & 7;
    int b    = blk >> 3;
    int r0   = (tile >> 2) * 16;
    int d0   = (tile & 3) * 16;

    int lane = threadIdx.x & 31;
    int half = lane >> 4;           // 0: K={0,1}, 1: K={2,3}
    int l16  = lane & 15;
    int kk0  = half * 2;

    const float* __restrict__ Hb = H + (size_t)b * NDIM * DDIM;

    v8f acc = {};
#pragma unroll 4
    for (int k = 0; k < NDIM; k += 4) {
        v2f af, bf;
        // A = Q^T : A[m=r][kk=n] = Q[k+kk][r0+m]
        af.x = Q[(k + kk0) * RDIM + r0 + l16];
        af.y = Q[(k + kk0 + 1) * RDIM + r0 + l16];
        // B : B[kk=n][nn=d] = H[b][k+kk][d0+nn]
        bf.x = Hb[(k + kk0) * DDIM + d0 + l16];
        bf.y = Hb[(k + kk0 + 1) * DDIM + d0 + l16];
        acc = __builtin_amdgcn_wmma_f32_16x16x4_f32(false, af, false, bf,
                                                    (short)0, acc, false, false);
    }
    // C/D layout: VGPR v -> M = v + half*8 (=r), N = l16 (=d)
    float* __restrict__ Ub = U + (size_t)b * RDIM * DDIM;
#pragma unroll
    for (int v = 0; v < 8; ++v)
        Ub[(r0 + half * 8 + v) * DDIM + d0 + l16] = acc[v];
}

// ---------------- Kernel 3: ctx + MLP -> s (B,O,r) ------------------------
__global__ __launch_bounds__(HIDD) void mlp_kernel(const float* __restrict__ ts_out,
                                                   const float* __restrict__ U,
                                                   const float* __restrict__ W1,
                                                   const float* __restrict__ b1,
                                                   const float* __restrict__ W2,
                                                   const float* __restrict__ b2,
                                                   float* __restrict__ sG) {
    int bo = blockIdx.x;                 // 0..383
    int o  = bo % ODIM;
    int b  = bo / ODIM;
    __shared__ float feat[64];           // [0..30]=ts, [31..62]=ctx
    __shared__ float hbuf[HIDD];
    int t = threadIdx.x;

    if (t < TDIM) feat[t] = ts_out[(b * ODIM + o) * TDIM + t];
    if (t >= 32 && t < 64) {             // ctx for r = t-32
        int r = t - 32;
        const float* Ur = U + ((size_t)b * RDIM + r) * DDIM;
        float ss = 0.0f;
#pragma unroll 8
        for (int d = 0; d < DDIM; ++d) { float u = Ur[d]; ss = fmaf(u, u, ss); }
        feat[TDIM + r] = sqrtf(ss * (1.0f / DDIM) + 1e-6f);
    }
    __syncthreads();

    float acc = b1[t];
    for (int j = 0; j < TDIM + RDIM; ++j)
        acc = fmaf(feat[j], W1[j * HIDD + t], acc);
    // exact gelu
    hbuf[t] = 0.5f * acc * (1.0f + erff(acc * 0.70710678118654752f));
    __syncthreads();

    if (t < RDIM) {
        float a2 = b2[t];
        for (int j = 0; j < HIDD; ++j)
            a2 = fmaf(hbuf[j], W2[j * RDIM + t], a2);
        sG[(b * ODIM + o) * RDIM + t] = softplus_f(a2);
    }
}

// ---------------- Kernel 4: out = (1-a)H + a * P @ (s.U)  -----------------
// Transposed tile: C^T[d][n] = sum_r V^T[d][r] * P^T[r][n], V = s (.) U[b]
// C/D layout then gives each lane 8 contiguous d-values of one output row.
__global__ __launch_bounds__(256) void out_kernel(const float* __restrict__ H,
                                                  const float* __restrict__ P,
                                                  const float* __restrict__ U,
                                                  const float* __restrict__ sG,
                                                  const float* __restrict__ alpha,
                                                  float* __restrict__ out) {
    // grid: BATCH*ODIM*(NDIM/128) = 6144 workgroups of 256 threads (8 waves)
    int blk    = blockIdx.x;
    int nchunk = blk & 15;
    int bo     = blk >> 4;
    int o      = bo % ODIM;
    int b      = bo / ODIM;
    int n0wg   = nchunk * 128;

    __shared__ float Vlds[RDIM * DDIM];    // 8 KB:  V[r][d] = s[r]*U[b][r][d]
    __shared__ float Plds[128 * RDIM];     // 16 KB: P rows n0wg..n0wg+127 (contiguous)

    int tid = threadIdx.x;
    const float* __restrict__ Ub = U + (size_t)b * RDIM * DDIM;
    const float* __restrict__ sb = sG + (b * ODIM + o) * RDIM;
    for (int i = tid; i < RDIM * DDIM; i += 256)
        Vlds[i] = sb[i >> 6] * Ub[i];

    const float* __restrict__ Pt = P + (size_t)n0wg * RDIM;   // 4096 contiguous floats
    {
        const v4f* src = (const v4f*)Pt;
        v4f*       dst = (v4f*)Plds;
        for (int i = tid; i < (128 * RDIM) / 4; i += 256) dst[i] = src[i];
    }
    __syncthreads();

    int wave = tid >> 5;
    int lane = tid & 31;
    int n0   = wave * 16;               // n-tile inside workgroup chunk
    int half = lane >> 4;
    int l16  = lane & 15;
    int kk0  = half * 2;

    v8f acc0 = {}, acc1 = {}, acc2 = {}, acc3 = {};
#pragma unroll
    for (int k = 0; k < RDIM; k += 4) {
        // B = P^T : B[kk=r][nn=n] = P[n0+nn][k+kk]
        v2f bf;
        bf.x = Plds[(n0 + l16) * RDIM + k + kk0];
        bf.y = Plds[(n0 + l16) * RDIM + k + kk0 + 1];
        // A = V^T : A[m=d][kk=r] = V[k+kk][d0+m]
        v2f a0, a1, a2, a3;
        a0.x = Vlds[(k + kk0) * DDIM + 0  + l16]; a0.y = Vlds[(k + kk0 + 1) * DDIM + 0  + l16];
        a1.x = Vlds[(k + kk0) * DDIM + 16 + l16]; a1.y = Vlds[(k + kk0 + 1) * DDIM + 16 + l16];
        a2.x = Vlds[(k + kk0) * DDIM + 32 + l16]; a2.y = Vlds[(k + kk0 + 1) * DDIM + 32 + l16];
        a3.x = Vlds[(k + kk0) * DDIM + 48 + l16]; a3.y = Vlds[(k + kk0 + 1) * DDIM + 48 + l16];
        acc0 = __builtin_amdgcn_wmma_f32_16x16x4_f32(false, a0, false, bf, (short)0, acc0, false, false);
        acc1 = __builtin_amdgcn_wmma_f32_16x16x4_f32(false, a1, false, bf, (short)0, acc1, false, false);
        acc2 = __builtin_amdgcn_wmma_f32_16x16x4_f32(false, a2, false, bf, (short)0, acc2, false, false);
        acc3 = __builtin_amdgcn_wmma_f32_16x16x4_f32(false, a3, false, bf, (short)0, acc3, false, false);
    }

    float a = alpha[0];
    a = fminf(fmaxf(a, 0.0f), 1.0f);

    int n = n0wg + n0 + l16;                       // this lane's output row
    int dbase = half * 8;                          // contiguous 8 d-values per acc
    const float* __restrict__ Hrow = H + ((size_t)b * NDIM + n) * DDIM;
    float* __restrict__ Orow = out + (((size_t)(b * ODIM + o)) * NDIM + n) * DDIM;

    v8f accs[4] = {acc0, acc1, acc2, acc3};
#pragma unroll
    for (int tIdx = 0; tIdx < 4; ++tIdx) {
        int d = tIdx * 16 + dbase;
        const v4f* h4 = (const v4f*)(Hrow + d);
        v4f h0 = h4[0], h1 = h4[1];
        v4f c0, c1;
#pragma unroll
        for (int v = 0; v < 4; ++v) { c0[v] = accs[tIdx][v]; c1[v] = accs[tIdx][v + 4]; }
        v4f r0 = h0 + a * (c0 - h0);
        v4f r1 = h1 + a * (c1 - h1);
        __builtin_nontemporal_store(r0, (v4f*)(Orow + d));
        __builtin_nontemporal_store(r1, (v4f*)(Orow + d) + 1);
    }
}

extern "C" void kernel_launch(void* const* d_in, const int* in_sizes, int n_in,
                              void* d_out, int out_size, void* d_ws, size_t ws_size,
                              hipStream_t stream) {
    const float* H    = (const float*)d_in[0];
    const float* ts   = (const float*)d_in[1];
    // d_in[2] = O (scalar int), compile-time constant here
    const float* Praw = (const float*)d_in[3];
    const float* Qraw = (const float*)d_in[4];
    const float* W1   = (const float*)d_in[5];
    const float* b1   = (const float*)d_in[6];
    const float* W2   = (const float*)d_in[7];
    const float* b2   = (const float*)d_in[8];
    const float* alp  = (const float*)d_in[9];
    float* out = (float*)d_out;

    float* ws = (float*)d_ws;
    float* Pw = ws;                        // 65536
    float* Qw = ws + 65536;                // 65536
    float* Uw = ws + 131072;               // 32*32*64 = 65536
    float* sw = ws + 196608;               // 32*12*32 = 12288

    sp_kernel<<<(NDIM * RDIM + 255) / 256, 256, 0, stream>>>(Praw, Pw, NDIM * RDIM);
    sp_kernel<<<(NDIM * RDIM + 255) / 256, 256, 0, stream>>>(Qraw, Qw, NDIM * RDIM);
    u_kernel<<<BATCH * 8, 32, 0, stream>>>(H, Qw, Uw);
    mlp_kernel<<<BATCH * ODIM, HIDD, 0, stream>>>(ts, Uw, W1, b1, W2, b2, sw);
    out_kernel<<<BATCH * ODIM * (NDIM / 128), 256, 0, stream>>>(H, Pw, Uw, sw, alp, out);
}